// GCNNodeBAShapes_10333691314777
// MI455X (gfx1250) — compile-verified
//
#include <hip/hip_runtime.h>
#include <hip/hip_bf16.h>

#define N_NODES     50000
#define N_EDGES     500000
#define IN_FEATS    64
#define H_FEATS     128
#define NUM_CLASSES 4
#define NUM_GRAPHS  8

typedef __attribute__((ext_vector_type(2))) float v2f;
typedef __attribute__((ext_vector_type(4))) float v4f;
typedef __attribute__((ext_vector_type(8))) float v8f;

static inline int cdiv(long a, long b) { return (int)((a + b - 1) / b); }

// ---------------------------------------------------------------- fill zero
__global__ void fill_zero_kernel(float* __restrict__ p, long n) {
    long i = (long)blockIdx.x * blockDim.x + threadIdx.x;
    if (i < n) p[i] = 0.0f;
}

// ---------------------------------------------------------------- degrees
__global__ void degree_kernel(const int* __restrict__ src, const int* __restrict__ dst,
                              float* __restrict__ deg_out, float* __restrict__ deg_in,
                              int n_edges) {
    int e = blockIdx.x * blockDim.x + threadIdx.x;
    if (e >= n_edges) return;
    atomicAdd(&deg_out[src[e]], 1.0f);
    atomicAdd(&deg_in[dst[e]], 1.0f);
}

// deg -> rsqrt(max(deg,1)) in place, over both arrays (contiguous)
__global__ void norm_kernel(float* __restrict__ p, long n) {
    long i = (long)blockIdx.x * blockDim.x + threadIdx.x;
    if (i >= n) return;
    float v = p[i];
    v = v < 1.0f ? 1.0f : v;
    p[i] = rsqrtf(v);
}

// ---------------------------------------------------------------- SpMM
// agg[dst] += e_w * out_norm[src] * x[src], edge x float4-chunk parallel
__global__ void spmm_kernel(const float* __restrict__ x, const int* __restrict__ src,
                            const int* __restrict__ dst, const float* __restrict__ ew,
                            const float* __restrict__ norm_out, float* __restrict__ agg,
                            int n_edges, int feat) {
    int chunks = feat >> 2;
    long tid = (long)blockIdx.x * blockDim.x + threadIdx.x;
    long total = (long)n_edges * chunks;
    if (tid >= total) return;
    int e = (int)(tid / chunks);
    int c = (int)(tid % chunks);
    int s = src[e], d = dst[e];
    float w = ew[e] * norm_out[s];
    v4f xv = *(const v4f*)&x[(long)s * feat + c * 4];
    float* o = &agg[(long)d * feat + c * 4];
    atomicAdd(o + 0, xv.x * w);
    atomicAdd(o + 1, xv.y * w);
    atomicAdd(o + 2, xv.z * w);
    atomicAdd(o + 3, xv.w * w);
}

// ---------------------------------------------------------------- dense GEMM (WMMA fp32)
// Y[rows x 128] = act( (X * in_norm_row)[rows x K] @ W[K x 128] + bias )
// One wave per 16-row block; 8 N-tiles of 16; K in steps of 4 via
// V_WMMA_F32_16X16X4_F32.
// W is staged in LDS *pair-interleaved*: sW2[(k>>1)*256 + col*2 + (k&1)],
// so the B operand {W[k][col], W[k+1][col]} (k always even at use sites)
// is one aligned ds_load_b64 directly into an even VGPR pair.
__global__ __launch_bounds__(256)
void gemm_wmma_kernel(const float* __restrict__ X, int ldx, int K,
                      const float* __restrict__ innorm,
                      const float* __restrict__ W, const float* __restrict__ bias,
                      float* __restrict__ Y, int rows, int do_relu) {
    extern __shared__ float sW2[];  // K * 128 floats, pair-interleaved
    for (int idx = threadIdx.x; idx < K * 128; idx += blockDim.x) {
        int k   = idx >> 7;        // idx / 128
        int col = idx & 127;       // idx % 128
        sW2[(k >> 1) * 256 + col * 2 + (k & 1)] = W[idx];
    }
    __syncthreads();

    int wave = threadIdx.x >> 5;
    int lane = threadIdx.x & 31;
    int wid  = blockIdx.x * (blockDim.x >> 5) + wave;
    int row_base = wid * 16;
    if (row_base >= rows) return;  // uniform per wave: EXEC stays all-ones

    int m    = lane & 15;   // A: row within tile / B,D: column within tile
    int half = lane >> 4;   // A/B: K offset (+2) ; D: row offset (+8)

    int row = row_base + m;
    float scale = innorm[row];
    const float* xrow = &X[(long)row * ldx];

    v8f acc[8];
#pragma unroll
    for (int t = 0; t < 8; ++t) acc[t] = (v8f){0,0,0,0,0,0,0,0};

    for (int k0 = 0; k0 < K; k0 += 4) {
        // A tile element pair: K = {k0+2*half, k0+2*half+1}
        v2f a = *(const v2f*)&xrow[k0 + 2 * half];
        a = a * scale;
        int pbase = ((k0 + 2 * half) >> 1) * 256;  // pair row in sW2
#pragma unroll
        for (int nt = 0; nt < 8; ++nt) {
            int col = nt * 16 + m;
            v2f b = *(const v2f*)&sW2[pbase + col * 2];  // {W[k][col], W[k+1][col]}
            acc[nt] = __builtin_amdgcn_wmma_f32_16x16x4_f32(
                false, a, false, b, (short)0, acc[nt], false, false);
        }
    }

    // store D: element [M = r + 8*half][N = nt*16 + m]
#pragma unroll
    for (int nt = 0; nt < 8; ++nt) {
        int col = nt * 16 + m;
        float bb = bias[col];
#pragma unroll
        for (int r = 0; r < 8; ++r) {
            int orow = row_base + r + 8 * half;
            float v = acc[nt][r] + bb;
            if (do_relu) v = fmaxf(v, 0.0f);
            Y[(long)orow * 128 + col] = v;
        }
    }
}

// ---------------------------------------------------------------- target ids
__global__ void tgt_kernel(const int* __restrict__ target_node,
                           const int* __restrict__ bnn, int* __restrict__ tgt) {
    if (threadIdx.x == 0 && blockIdx.x == 0) {
        int off = 0;
        for (int g = 0; g < NUM_GRAPHS; ++g) {
            tgt[g] = target_node[g] + off;
            off += bnn[g];
        }
    }
}

// ---------------------------------------------------------------- layer-3 sparse agg (targets only)
__global__ void target_agg_kernel(const float* __restrict__ h, const int* __restrict__ src,
                                  const int* __restrict__ dst, const float* __restrict__ ew,
                                  const float* __restrict__ norm_out,
                                  const int* __restrict__ tgt, float* __restrict__ aggt,
                                  int n_edges) {
    int e = blockIdx.x * blockDim.x + threadIdx.x;
    if (e >= n_edges) return;
    int d = dst[e];
#pragma unroll
    for (int g = 0; g < NUM_GRAPHS; ++g) {
        if (d == tgt[g]) {
            int s = src[e];
            float w = ew[e] * norm_out[s];
            const float* rowp = &h[(long)s * H_FEATS];
            for (int k = 0; k < H_FEATS; ++k)
                atomicAdd(&aggt[g * H_FEATS + k], w * rowp[k]);
        }
    }
}

// ---------------------------------------------------------------- final 8x4 head
__global__ void head_kernel(const float* __restrict__ aggt, const float* __restrict__ norm_in,
                            const int* __restrict__ tgt, const float* __restrict__ W3,
                            const float* __restrict__ b3, float* __restrict__ out) {
    int t = threadIdx.x;
    if (t >= NUM_GRAPHS * NUM_CLASSES) return;
    int g = t >> 2, c = t & 3;
    float nin = norm_in[tgt[g]];
    float acc = b3[c];
    for (int k = 0; k < H_FEATS; ++k)
        acc += aggt[g * H_FEATS + k] * nin * W3[k * NUM_CLASSES + c];
    out[g * NUM_CLASSES + c] = acc;
}

// ================================================================ launch
extern "C" void kernel_launch(void* const* d_in, const int* in_sizes, int n_in,
                              void* d_out, int out_size, void* d_ws, size_t ws_size,
                              hipStream_t stream) {
    const float* in_feat = (const float*)d_in[0];
    const float* ew      = (const float*)d_in[1];
    const int*   src     = (const int*)d_in[2];
    const int*   dst     = (const int*)d_in[3];
    const int*   bnn     = (const int*)d_in[4];
    const int*   tnode   = (const int*)d_in[5];
    const float* W1 = (const float*)d_in[6];
    const float* b1 = (const float*)d_in[7];
    const float* W2 = (const float*)d_in[8];
    const float* b2 = (const float*)d_in[9];
    const float* W3 = (const float*)d_in[10];
    const float* b3 = (const float*)d_in[11];
    float* out = (float*)d_out;

    float* ws       = (float*)d_ws;
    float* norm_out = ws;                                   // N_NODES
    float* norm_in  = ws + N_NODES;                         // N_NODES
    float* agg      = ws + 2L * N_NODES;                    // N_NODES * 128
    float* h        = agg + (long)N_NODES * H_FEATS;        // N_NODES * 128
    float* aggt     = h + (long)N_NODES * H_FEATS;          // 8 * 128
    int*   tgt      = (int*)(aggt + NUM_GRAPHS * H_FEATS);  // 8 ints

    const int BT = 256;

    // degrees -> norms
    fill_zero_kernel<<<cdiv(2L * N_NODES, BT), BT, 0, stream>>>(norm_out, 2L * N_NODES);
    degree_kernel<<<cdiv(N_EDGES, BT), BT, 0, stream>>>(src, dst, norm_out, norm_in, N_EDGES);
    norm_kernel<<<cdiv(2L * N_NODES, BT), BT, 0, stream>>>(norm_out, 2L * N_NODES);

    // layer 1: SpMM (feat=64) + WMMA GEMM 64->128 + ReLU
    fill_zero_kernel<<<cdiv((long)N_NODES * IN_FEATS, BT), BT, 0, stream>>>(agg, (long)N_NODES * IN_FEATS);
    spmm_kernel<<<cdiv((long)N_EDGES * (IN_FEATS / 4), BT), BT, 0, stream>>>(
        in_feat, src, dst, ew, norm_out, agg, N_EDGES, IN_FEATS);
    {
        int waves = N_NODES / 16;                 // 3125
        int blocks = cdiv(waves, BT / 32);
        gemm_wmma_kernel<<<blocks, BT, IN_FEATS * 128 * sizeof(float), stream>>>(
            agg, IN_FEATS, IN_FEATS, norm_in, W1, b1, h, N_NODES, 1);
    }

    // layer 2: SpMM (feat=128) + WMMA GEMM 128->128 + ReLU (h -> agg -> h)
    fill_zero_kernel<<<cdiv((long)N_NODES * H_FEATS, BT), BT, 0, stream>>>(agg, (long)N_NODES * H_FEATS);
    spmm_kernel<<<cdiv((long)N_EDGES * (H_FEATS / 4), BT), BT, 0, stream>>>(
        h, src, dst, ew, norm_out, agg, N_EDGES, H_FEATS);
    {
        int waves = N_NODES / 16;
        int blocks = cdiv(waves, BT / 32);
        gemm_wmma_kernel<<<blocks, BT, H_FEATS * 128 * sizeof(float), stream>>>(
            agg, H_FEATS, H_FEATS, norm_in, W2, b2, h, N_NODES, 1);
    }

    // layer 3: only 8 target rows needed
    fill_zero_kernel<<<cdiv((long)NUM_GRAPHS * H_FEATS, BT), BT, 0, stream>>>(
        aggt, (long)NUM_GRAPHS * H_FEATS);
    tgt_kernel<<<1, 32, 0, stream>>>(tnode, bnn, tgt);
    target_agg_kernel<<<cdiv(N_EDGES, BT), BT, 0, stream>>>(
        h, src, dst, ew, norm_out, tgt, aggt, N_EDGES);
    head_kernel<<<1, 32, 0, stream>>>(aggt, norm_in, tgt, W3, b3, out);
}